// GCNModel_57853209477141
// MI455X (gfx1250) — compile-verified
//
#include <hip/hip_runtime.h>
#include <hip/hip_bf16.h>

typedef __attribute__((ext_vector_type(2))) float v2f;
typedef __attribute__((ext_vector_type(8))) float v8f;

#define HID    128
#define QDIM   768
#define PROCQ  128
#define OUTCH  32

// ---------------------------------------------------------------------------
// Degree / normalization kernels
// ---------------------------------------------------------------------------
__global__ void k_fill_ones(float* __restrict__ p, int n) {
    int i = blockIdx.x * blockDim.x + threadIdx.x;
    if (i < n) p[i] = 1.0f;            // self-loop contributes 1 to every node
}

__global__ void k_deg_accum(float* __restrict__ deg, const int* __restrict__ dst, int E) {
    int i = blockIdx.x * blockDim.x + threadIdx.x;
    if (i < E) atomicAdd(&deg[dst[i]], 1.0f);
}

__global__ void k_make_dinv(float* __restrict__ deg, int n) {   // in-place deg -> 1/sqrt(deg)
    int i = blockIdx.x * blockDim.x + threadIdx.x;
    if (i < n) {
        float d = deg[i];
        deg[i] = d > 0.0f ? rsqrtf(d) : 0.0f;
    }
}

// ---------------------------------------------------------------------------
// Question head: q = relu(Q @ fc0_w + fc0_b)   [64 x 768] @ [768 x 128]
// Tiny (12.6 MFLOP) -> simple scalar kernel, one block per graph.
// ---------------------------------------------------------------------------
__global__ void k_q_head(const float* __restrict__ Q, const float* __restrict__ W,
                         const float* __restrict__ b, float* __restrict__ q) {
    int j = threadIdx.x;        // 0..127
    int g = blockIdx.x;         // 0..63
    float acc = b[j];
    const float* qr = Q + (size_t)g * QDIM;
    for (int k = 0; k < QDIM; ++k)
        acc = fmaf(qr[k], W[(size_t)k * PROCQ + j], acc);
    q[(size_t)g * PROCQ + j] = fmaxf(acc, 0.0f);
}

// ---------------------------------------------------------------------------
// WMMA fp32 GEMM:  out = A[M x K] @ W[K x NOUT]  (+ epilogue)
//   one wave computes a 16-row x NOUT tile, 8 waves/block -> 128 rows/block
//   W staged through LDS in 128-row chunks; v_wmma_f32_16x16x4_f32 inner op.
//
// A-matrix 16x4 f32 layout (ISA): lanes 0-15 hold K=0,1; lanes 16-31 K=2,3
// B-matrix 4x16 f32 layout: VGPR j, halfsel s -> row K = 2*j + s ... so a lane
//   needs the pair (W[g+s][n], W[g+s+2][n]) for k-group g.  We stage W into
//   LDS pre-paired as float2 so the whole B operand is ONE ds_load_b64:
//     Wl[(g/4*2 + s)*NOUT + n] = { W[g+s][n], W[g+s+2][n] }
// C/D 16x16 f32 layout: VGPR i -> M=i (lanes 0-15), M=i+8 (lanes 16-31)
//
// EPI 0: v *= dinv[row];  store to out0 (hs) AND out1 (agg, self-loop init)
// EPI 1: v = relu(v + bias[col]); store out0
// EPI 2: v = v + bias[col];       store out0
// CONCAT: A column c<128 from A (h), else from Qm[batch[row]] (c-128)
// ---------------------------------------------------------------------------
template<int K, int NOUT, bool CONCAT, int EPI>
__global__ __launch_bounds__(256)
void k_gemm_wmma(const float* __restrict__ A,
                 const float* __restrict__ Qm, const int* __restrict__ batch,
                 const float* __restrict__ W,  const float* __restrict__ bias,
                 const float* __restrict__ dinv,
                 float* __restrict__ out0, float* __restrict__ out1, int M) {
    constexpr int KC = 128;
    constexpr int NT = NOUT / 16;
    __shared__ float2 Wl[(KC / 2) * NOUT];   // pair-interleaved B tile

    const int tid     = threadIdx.x;
    const int lane    = tid & 31;
    const int wave    = tid >> 5;
    const int halfsel = lane >> 4;      // 0 or 1
    const int l15     = lane & 15;
    const int mbase   = blockIdx.x * 128 + wave * 16;

    int arow = mbase + l15;
    if (arow >= M) arow = M - 1;        // clamp: keep EXEC all-1s for WMMA

    const float* qrowp = nullptr;
    if constexpr (CONCAT) qrowp = Qm + (size_t)batch[arow] * PROCQ;

    v8f acc[NT];
#pragma unroll
    for (int t = 0; t < NT; ++t)
        acc[t] = (v8f){0.f, 0.f, 0.f, 0.f, 0.f, 0.f, 0.f, 0.f};

    for (int kc = 0; kc < K; kc += KC) {
        __syncthreads();
        // cooperative load of the W chunk into LDS, pair-interleaved:
        // global row r (in chunk), col n  ->  Wl[(r/4*2 + (r&1))*NOUT + n].{x|y}
        // where .x for (r%4)<2, .y for (r%4)>=2
        const float* Wc = W + (size_t)kc * NOUT;
        for (int i = tid * 4; i < KC * NOUT; i += 256 * 4) {
            const float4 v = *reinterpret_cast<const float4*>(&Wc[i]);
            const int r  = i / NOUT;
            const int n0 = i % NOUT;
            const int pr = ((r >> 2) * 2 + (r & 1)) * NOUT + n0;  // pair-row base
            if ((r & 2) == 0) {
                Wl[pr + 0].x = v.x; Wl[pr + 1].x = v.y;
                Wl[pr + 2].x = v.z; Wl[pr + 3].x = v.w;
            } else {
                Wl[pr + 0].y = v.x; Wl[pr + 1].y = v.y;
                Wl[pr + 2].y = v.z; Wl[pr + 3].y = v.w;
            }
        }
        __syncthreads();

        for (int k = 0; k < KC; k += 4) {
            const int c = kc + k + 2 * halfsel;   // even -> 8B-aligned pair
            v2f a;
            if constexpr (CONCAT) {
                if (c < HID) {
                    a.x = A[(size_t)arow * HID + c];
                    a.y = A[(size_t)arow * HID + c + 1];
                } else {
                    a.x = qrowp[c - HID];
                    a.y = qrowp[c - HID + 1];
                }
            } else {
                const float2 t = *reinterpret_cast<const float2*>(A + (size_t)arow * K + c);
                a.x = t.x; a.y = t.y;
            }
            const int pbase = ((k >> 2) * 2 + halfsel) * NOUT + l15;
#pragma unroll
            for (int nt = 0; nt < NT; ++nt) {
                const float2 bb = Wl[pbase + nt * 16];   // one ds_load_b64
                v2f b;
                b.x = bb.x;          // W[k+halfsel][col]
                b.y = bb.y;          // W[k+2+halfsel][col]
                acc[nt] = __builtin_amdgcn_wmma_f32_16x16x4_f32(
                    false, a, false, b, (short)0, acc[nt], false, false);
            }
        }
    }

    // epilogue
#pragma unroll
    for (int nt = 0; nt < NT; ++nt) {
#pragma unroll
        for (int i = 0; i < 8; ++i) {
            const int row = mbase + i + 8 * halfsel;
            const int col = nt * 16 + l15;
            if (row < M) {
                float v = acc[nt][i];
                if constexpr (EPI == 0) {
                    v *= dinv[row];
                    out0[(size_t)row * NOUT + col] = v;   // hs
                    out1[(size_t)row * NOUT + col] = v;   // agg init (self loop)
                } else if constexpr (EPI == 1) {
                    v = fmaxf(v + bias[col], 0.0f);
                    out0[(size_t)row * NOUT + col] = v;
                } else {
                    out0[(size_t)row * NOUT + col] = v + bias[col];
                }
            }
        }
    }
}

// ---------------------------------------------------------------------------
// Edge scatter: agg[dst] += hs[src], one wave per edge, float4 per lane.
// hs/agg (25.6 MB each) fit in the 192 MB L2 -> atomics resolve in-cache.
// ---------------------------------------------------------------------------
__global__ __launch_bounds__(256)
void k_scatter_edges(const float* __restrict__ hs, float* __restrict__ agg,
                     const int* __restrict__ src, const int* __restrict__ dst, int E) {
    const int w    = (int)((blockIdx.x * 256u + threadIdx.x) >> 5);
    const int lane = threadIdx.x & 31;
    if (w >= E) return;
    const int s = src[w];
    const int d = dst[w];
    const float4 v = *reinterpret_cast<const float4*>(hs + (size_t)s * HID + lane * 4);
    float* p = agg + (size_t)d * HID + lane * 4;
    atomicAdd(p + 0, v.x);
    atomicAdd(p + 1, v.y);
    atomicAdd(p + 2, v.z);
    atomicAdd(p + 3, v.w);
}

// ---------------------------------------------------------------------------
// Finalize GCN layer: h = relu(agg * dinv[row] + bias)
// ---------------------------------------------------------------------------
__global__ void k_gcn_finalize(const float* __restrict__ agg, const float* __restrict__ dinv,
                               const float* __restrict__ bias, float* __restrict__ out, int N) {
    const int i = blockIdx.x * blockDim.x + threadIdx.x;   // over N * (HID/4)
    if (i >= N * (HID / 4)) return;
    const int row = i >> 5;
    const int c4  = (i & 31) * 4;
    const float s = dinv[row];
    const float4 v = *reinterpret_cast<const float4*>(agg  + (size_t)row * HID + c4);
    const float4 b = *reinterpret_cast<const float4*>(bias + c4);
    float4 o;
    o.x = fmaxf(fmaf(v.x, s, b.x), 0.0f);
    o.y = fmaxf(fmaf(v.y, s, b.y), 0.0f);
    o.z = fmaxf(fmaf(v.z, s, b.z), 0.0f);
    o.w = fmaxf(fmaf(v.w, s, b.w), 0.0f);
    *reinterpret_cast<float4*>(out + (size_t)row * HID + c4) = o;
}

// ---------------------------------------------------------------------------
// Launch
// ---------------------------------------------------------------------------
extern "C" void kernel_launch(void* const* d_in, const int* in_sizes, int n_in,
                              void* d_out, int out_size, void* d_ws, size_t ws_size,
                              hipStream_t stream) {
    const float* x     = (const float*)d_in[0];
    const int*   ei    = (const int*)  d_in[1];   // [2, E]: row0 = src, row1 = dst
    const int*   batch = (const int*)  d_in[2];
    const float* qemb  = (const float*)d_in[3];
    const float* W0    = (const float*)d_in[4];
    const float* b0    = (const float*)d_in[5];
    const float* W1    = (const float*)d_in[6];
    const float* b1    = (const float*)d_in[7];
    const float* W2    = (const float*)d_in[8];
    const float* b2    = (const float*)d_in[9];
    const float* fc0w  = (const float*)d_in[10];
    const float* fc0b  = (const float*)d_in[11];
    const float* fc1w  = (const float*)d_in[12];
    const float* fc1b  = (const float*)d_in[13];
    const float* fc2w  = (const float*)d_in[14];
    const float* fc2b  = (const float*)d_in[15];

    const int N = in_sizes[0] / HID;      // 50000
    const int E = in_sizes[1] / 2;        // 800000
    const int G = in_sizes[3] / QDIM;     // 64

    const int* src = ei;
    const int* dst = ei + E;

    // workspace layout (floats)
    float* dinv = (float*)d_ws;                    // N
    float* qmat = dinv + (size_t)N;                // G * PROCQ
    float* B0   = qmat + (size_t)G * PROCQ;        // N * HID  (hs)
    float* B1   = B0   + (size_t)N * HID;          // N * HID  (agg)
    float* B2   = B1   + (size_t)N * HID;          // N * HID  (h)

    const int T = 256;
    const int gN   = (N + T - 1) / T;
    const int gE   = (E + T - 1) / T;
    const int gGB  = (N + 127) / 128;              // gemm blocks (128 rows each)
    const int gSC  = (E + 7) / 8;                  // scatter: 8 waves/block
    const int gFIN = (N * (HID / 4) + T - 1) / T;

    // 1) GCN normalization: deg = 1 + in-degree; dinv = rsqrt(deg)
    k_fill_ones<<<gN, T, 0, stream>>>(dinv, N);
    k_deg_accum<<<gE, T, 0, stream>>>(dinv, dst, E);
    k_make_dinv<<<gN, T, 0, stream>>>(dinv, N);

    // 2) question head
    k_q_head<<<G, PROCQ, 0, stream>>>(qemb, fc0w, fc0b, qmat);

    // 3) GCN layer 0:  hs = (x@W0)*dinv ; agg = hs ; agg += edges ; h = relu(agg*dinv + b0)
    k_gemm_wmma<HID, HID, false, 0><<<gGB, T, 0, stream>>>(
        x, nullptr, nullptr, W0, nullptr, dinv, B0, B1, N);
    k_scatter_edges<<<gSC, T, 0, stream>>>(B0, B1, src, dst, E);
    k_gcn_finalize<<<gFIN, T, 0, stream>>>(B1, dinv, b0, B2, N);

    // 4) GCN layer 1
    k_gemm_wmma<HID, HID, false, 0><<<gGB, T, 0, stream>>>(
        B2, nullptr, nullptr, W1, nullptr, dinv, B0, B1, N);
    k_scatter_edges<<<gSC, T, 0, stream>>>(B0, B1, src, dst, E);
    k_gcn_finalize<<<gFIN, T, 0, stream>>>(B1, dinv, b1, B2, N);

    // 5) GCN layer 2
    k_gemm_wmma<HID, HID, false, 0><<<gGB, T, 0, stream>>>(
        B2, nullptr, nullptr, W2, nullptr, dinv, B0, B1, N);
    k_scatter_edges<<<gSC, T, 0, stream>>>(B0, B1, src, dst, E);
    k_gcn_finalize<<<gFIN, T, 0, stream>>>(B1, dinv, b2, B2, N);

    // 6) fc1: out = relu(concat(h, q[batch]) @ fc1_w + fc1_b)   (K = 256)
    k_gemm_wmma<HID + PROCQ, HID, true, 1><<<gGB, T, 0, stream>>>(
        B2, qmat, batch, fc1w, fc1b, nullptr, B0, nullptr, N);

    // 7) fc2: d_out = fc1_out @ fc2_w + fc2_b   (NOUT = 32)
    k_gemm_wmma<HID, OUTCH, false, 2><<<gGB, T, 0, stream>>>(
        B0, nullptr, nullptr, fc2w, fc2b, nullptr, (float*)d_out, nullptr, N);
}